// InterLoss_29111288332477
// MI455X (gfx1250) — compile-verified
//
#include <hip/hip_runtime.h>
#include <hip/hip_bf16.h>

#define NUM_CLASS 1000
#define FEAT_DIM  512
#define BATCH     32768
#define THRESHOLD 5.0f
#define TILES     63                 // ceil(1000/16)
#define NTILE2    (TILES * TILES)    // 3969 output tiles

typedef float v2f __attribute__((ext_vector_type(2)));
typedef float v8f __attribute__((ext_vector_type(8)));

// workspace layout (floats):
//   [0]            loss accumulator
//   [1 .. 1000]    per-class counts
//   [1024 .. 2023] per-row squared norms of new_center
#define WS_LOSS 0
#define WS_CNT  1
#define WS_SQ   1024

// ---- 1) zero output center region + workspace scalars (re-run every call) ----
__global__ void il_zero_kernel(float* __restrict__ out_center, float* __restrict__ wsf) {
    int idx = blockIdx.x * blockDim.x + threadIdx.x;
    if (idx < NUM_CLASS * FEAT_DIM) out_center[idx] = 0.0f;
    if (idx < 2048)                 wsf[idx]        = 0.0f;
}

// ---- 2) per-class counts ----
__global__ void il_count_kernel(const int* __restrict__ labels, float* __restrict__ wsf) {
    int s = blockIdx.x * blockDim.x + threadIdx.x;
    if (s < BATCH) {
        unsafeAtomicAdd(&wsf[WS_CNT + labels[s]], 1.0f);
    }
}

// ---- 3) segment-sum features into out_center via f32 global atomics ----
// one thread = one float4 of one sample (128 float4 per sample)
__global__ void il_scatter_kernel(const float* __restrict__ features,
                                  const int* __restrict__ labels,
                                  float* __restrict__ out_center) {
    int idx = blockIdx.x * blockDim.x + threadIdx.x;   // over BATCH*FEAT_DIM/4
    int s  = idx >> 7;
    int k4 = (idx & 127) << 2;
    float4 v = reinterpret_cast<const float4*>(features)[idx];
    int c = labels[s];
    float* dst = out_center + c * FEAT_DIM + k4;
    unsafeAtomicAdd(dst + 0, v.x);
    unsafeAtomicAdd(dst + 1, v.y);
    unsafeAtomicAdd(dst + 2, v.z);
    unsafeAtomicAdd(dst + 3, v.w);
}

// ---- 4) new_center = center + sums/max(cnt,1); accumulate row squared norms ----
__global__ void il_center_kernel(const float* __restrict__ center,
                                 float* __restrict__ out_center,
                                 float* __restrict__ wsf) {
    int idx = blockIdx.x * blockDim.x + threadIdx.x;
    if (idx >= NUM_CLASS * FEAT_DIM) return;
    int c = idx >> 9;                                  // FEAT_DIM == 512
    float cnt = fmaxf(wsf[WS_CNT + c], 1.0f);
    float nc  = center[idx] + out_center[idx] / cnt;
    out_center[idx] = nc;
    unsafeAtomicAdd(&wsf[WS_SQ + c], nc * nc);
}

// ---- 5) Gram matrix via V_WMMA_F32_16X16X4_F32 + hinge loss reduction ----
// One wave per 16x16 tile of G = N*N^T; 8 waves per block.
// f32 A 16x4 layout: lane L holds A[M=L%16][K = vgpr + 2*(L/16)]
// f32 B 4x16 layout mirrors with N across lanes -> identical load addressing.
__global__ void __launch_bounds__(256)
il_gram_loss_kernel(const float* __restrict__ NC,      // new_center [1008-padded rows read clamped]
                    const float* __restrict__ wsf,
                    float* __restrict__ wsloss) {
    const int lane = threadIdx.x & 31;
    const int wave = threadIdx.x >> 5;
    const int tile = blockIdx.x * 8 + wave;

    float partial = 0.0f;
    if (tile < NTILE2) {                 // wave-uniform branch: EXEC stays all-ones
        const int it = tile / TILES;
        const int jt = tile % TILES;
        const int m  = lane & 15;
        const int kh = lane >> 4;        // 0 or 1 -> K offset 2*kh

        const int rowA = it * 16 + m;
        const int rowB = jt * 16 + m;
        const float maskA = (rowA < NUM_CLASS) ? 1.0f : 0.0f;
        const float maskB = (rowB < NUM_CLASS) ? 1.0f : 0.0f;
        const int rA = (rowA < NUM_CLASS) ? rowA : (NUM_CLASS - 1);
        const int rB = (rowB < NUM_CLASS) ? rowB : (NUM_CLASS - 1);
        const float* pa = NC + rA * FEAT_DIM + 2 * kh;
        const float* pb = NC + rB * FEAT_DIM + 2 * kh;

        v8f acc = {};
        #pragma unroll 4
        for (int k0 = 0; k0 < FEAT_DIM; k0 += 4) {
            v2f a = *reinterpret_cast<const v2f*>(pa + k0);
            v2f b = *reinterpret_cast<const v2f*>(pb + k0);
            a = a * maskA;               // zero-pad rows >= 1000 without diverging EXEC
            b = b * maskB;
            acc = __builtin_amdgcn_wmma_f32_16x16x4_f32(
                /*neg_a=*/false, a, /*neg_b=*/false, b,
                /*c_mod=*/(short)0, acc, /*reuse_a=*/false, /*reuse_b=*/false);
        }

        // C/D layout: vgpr r, lanes 0-15 -> M=r, lanes 16-31 -> M=r+8; N = lane%16
        const int jg = jt * 16 + m;
        const float sqj = (jg < NUM_CLASS) ? wsf[WS_SQ + jg] : 0.0f;
        #pragma unroll
        for (int r = 0; r < 8; ++r) {
            int ig = it * 16 + r + 8 * kh;
            if (ig < NUM_CLASS && jg < NUM_CLASS) {
                float g  = acc[r];
                float d2 = wsf[WS_SQ + ig] + sqj - 2.0f * g;
                float d  = sqrtf(fmaxf(d2, 0.0f));
                partial += (d < THRESHOLD) ? (THRESHOLD - d) : 0.0f;
            }
        }
    }

    // wave32 reduction, then one atomic per wave
    #pragma unroll
    for (int off = 16; off > 0; off >>= 1)
        partial += __shfl_xor(partial, off, 32);
    if ((threadIdx.x & 31) == 0)
        unsafeAtomicAdd(wsloss, partial);
}

// ---- 6) loss = sum / (1000*1000) ----
__global__ void il_finalize_kernel(const float* __restrict__ wsf, float* __restrict__ out) {
    if (threadIdx.x == 0 && blockIdx.x == 0)
        out[0] = wsf[WS_LOSS] * (1.0f / ((float)NUM_CLASS * (float)NUM_CLASS));
}

extern "C" void kernel_launch(void* const* d_in, const int* in_sizes, int n_in,
                              void* d_out, int out_size, void* d_ws, size_t ws_size,
                              hipStream_t stream) {
    const float* features = (const float*)d_in[0];   // [32768, 512] f32
    const int*   labels   = (const int*)d_in[1];     // [32768] int
    const float* center   = (const float*)d_in[2];   // [1000, 512] f32

    float* out        = (float*)d_out;               // [0]=loss, [1..512000]=new_center
    float* out_center = out + 1;
    float* wsf        = (float*)d_ws;

    il_zero_kernel<<<(NUM_CLASS * FEAT_DIM + 255) / 256, 256, 0, stream>>>(out_center, wsf);
    il_count_kernel<<<(BATCH + 255) / 256, 256, 0, stream>>>(labels, wsf);
    il_scatter_kernel<<<(BATCH * FEAT_DIM / 4) / 256, 256, 0, stream>>>(features, labels, out_center);
    il_center_kernel<<<(NUM_CLASS * FEAT_DIM + 255) / 256, 256, 0, stream>>>(center, out_center, wsf);
    il_gram_loss_kernel<<<(NTILE2 + 7) / 8, 256, 0, stream>>>(out_center, wsf, wsf + WS_LOSS);
    il_finalize_kernel<<<1, 1, 0, stream>>>(wsf, out);
}